// SimpleSSM_67379446940102
// MI455X (gfx1250) — compile-verified
//
#include <hip/hip_runtime.h>

// ---------------- problem constants (from reference) ----------------
#define HID   768
#define DIN   1536
#define NST   16
#define BSZ   4
#define SEQ   2048
#define NXZ   (2*DIN)       // 3072
#define NXP   (2*NST+1)     // 33
#define M_TOT (BSZ*SEQ)     // 8192

typedef __attribute__((ext_vector_type(2))) float v2f;
typedef __attribute__((ext_vector_type(4))) float v4f;
typedef __attribute__((ext_vector_type(8))) float v8f;

__device__ __forceinline__ float silu_f(float x)  { return x / (1.0f + expf(-x)); }
__device__ __forceinline__ float softplus_f(float x) {
    return (x > 20.0f) ? x : log1pf(expf(x));
}

// ====================================================================
// Register-blocked WMMA fp32 GEMM:  C[M x N] = A[M x K] * W[N x K]^T
// One wave computes a 32x64 macro-tile of C = 2 (M) x 4 (N) WMMA tiles.
// Per k-step: 2 A-fragment loads + 4 B-fragment loads feed 8
// V_WMMA_F32_16X16X4_F32 ops (vs 2 loads : 1 wmma unblocked).
//
// CDNA5 32-bit 16x4 A layout: lanes 0..15 hold K{0,1} in v0/v1 (M=lane),
// lanes 16..31 hold K{2,3} -> contiguous float2 at A[row][k + 2*(lane>>4)].
// B = W^T so B[k][n] = W[n][k]: same contiguous float2 on row-major W with
// n = lane&15. C/D: vgpr r -> M = r + 8*(lane>>4), N = lane&15.
// ====================================================================
template<int KDIM, int NDIM>
__global__ void wmma_gemm_nt(const float* __restrict__ A,
                             const float* __restrict__ W,
                             float* __restrict__ C)
{
    constexpr int NWT = NDIM / 64;             // N macro-tiles
    const int lane = threadIdx.x & 31;
    const int wave = threadIdx.x >> 5;
    const int wid  = blockIdx.x * 8 + wave;    // global wave-tile id
    const int mi   = wid / NWT;
    const int ni   = wid % NWT;
    const int m0   = mi * 32;
    const int n0   = ni * 64;

    const int row  = lane & 15;
    const int half = lane >> 4;                // 0: K{0,1}, 1: K{2,3}

    const float* __restrict__ a0 = A + (size_t)(m0 + row) * KDIM + 2 * half;
    const float* __restrict__ a1 = a0 + (size_t)16 * KDIM;
    const float* __restrict__ b0 = W + (size_t)(n0 + row) * KDIM + 2 * half;

    v8f acc[2][4] = {};

#pragma unroll 2
    for (int k = 0; k < KDIM; k += 4) {
        v2f av0 = *(const v2f*)(a0 + k);
        v2f av1 = *(const v2f*)(a1 + k);
        v2f bv[4];
#pragma unroll
        for (int j = 0; j < 4; ++j)
            bv[j] = *(const v2f*)(b0 + (size_t)j * 16 * KDIM + k);

#pragma unroll
        for (int j = 0; j < 4; ++j) {
            acc[0][j] = __builtin_amdgcn_wmma_f32_16x16x4_f32(
                            false, av0, false, bv[j], (short)0, acc[0][j], false, false);
            acc[1][j] = __builtin_amdgcn_wmma_f32_16x16x4_f32(
                            false, av1, false, bv[j], (short)0, acc[1][j], false, false);
        }
    }

    float* __restrict__ cbase = C + (size_t)m0 * NDIM + n0;
#pragma unroll
    for (int mm = 0; mm < 2; ++mm)
#pragma unroll
        for (int j = 0; j < 4; ++j)
#pragma unroll
            for (int r = 0; r < 8; ++r)
                cbase[(size_t)(16 * mm + 8 * half + r) * NDIM + 16 * j + row]
                    = acc[mm][j][r];
}

// ====================================================================
// Causal depthwise conv (K=4) over sequence + SiLU, float4 over channels.
// xi = xz[:, :, 0:DIN];  xc[b,s,d] = silu(sum_j w[d,j] * xi[b,s-3+j,d])
// ====================================================================
__global__ void conv_silu_kernel(const float* __restrict__ xz,   // [M_TOT][NXZ]
                                 const float* __restrict__ cw,   // [DIN][1][4]
                                 float* __restrict__ xc)         // [M_TOT][DIN]
{
    const long long idx = (long long)blockIdx.x * blockDim.x + threadIdx.x;
    if (idx >= (long long)M_TOT * (DIN / 4)) return;
    const int d4 = (int)(idx % (DIN / 4));     // group of 4 channels
    const long long ms = idx / (DIN / 4);      // b*SEQ + s
    const int s  = (int)(ms % SEQ);
    const int d  = d4 * 4;

    // weights for 4 consecutive channels: cw[d+c][j]
    v4f wc[4];
#pragma unroll
    for (int c = 0; c < 4; ++c) wc[c] = *(const v4f*)(cw + (size_t)(d + c) * 4);

    const long long base = ms * NXZ + d;       // xi lives in cols [0, DIN)
    v4f x0 = *(const v4f*)(xz + base);
    v4f acc;
#pragma unroll
    for (int c = 0; c < 4; ++c) acc[c] = wc[c].w * x0[c];
    if (s >= 1) { v4f v = *(const v4f*)(xz + base - 1LL * NXZ);
#pragma unroll
        for (int c = 0; c < 4; ++c) acc[c] += wc[c].z * v[c]; }
    if (s >= 2) { v4f v = *(const v4f*)(xz + base - 2LL * NXZ);
#pragma unroll
        for (int c = 0; c < 4; ++c) acc[c] += wc[c].y * v[c]; }
    if (s >= 3) { v4f v = *(const v4f*)(xz + base - 3LL * NXZ);
#pragma unroll
        for (int c = 0; c < 4; ++c) acc[c] += wc[c].x * v[c]; }

    v4f outv;
#pragma unroll
    for (int c = 0; c < 4; ++c) outv[c] = silu_f(acc[c]);
    *(v4f*)(xc + ms * DIN + d) = outv;
}

// ====================================================================
// xp[b,s,p] = dot(xc[b,s,:], W_xp[p,:]) — N=33 is too ragged for WMMA;
// one wave32 per (b,s,p) with float4 loads + shuffle reduction.
// ====================================================================
__global__ void xp_kernel(const float* __restrict__ xc,    // [M_TOT][DIN]
                          const float* __restrict__ Wxp,   // [NXP][DIN]
                          float* __restrict__ xp)          // [M_TOT][NXP]
{
    const long long gwave = ((long long)blockIdx.x * blockDim.x + threadIdx.x) >> 5;
    const int lane = threadIdx.x & 31;
    if (gwave >= (long long)M_TOT * NXP) return;
    const int p  = (int)(gwave % NXP);
    const long long ms = gwave / NXP;

    const v4f* __restrict__ xr = (const v4f*)(xc  + ms * DIN);
    const v4f* __restrict__ wr = (const v4f*)(Wxp + (size_t)p * DIN);

    float acc = 0.0f;
#pragma unroll
    for (int k = lane; k < DIN / 4; k += 32) {   // 12 iterations
        v4f a = xr[k], b = wr[k];
        acc += a.x * b.x + a.y * b.y + a.z * b.z + a.w * b.w;
    }
#pragma unroll
    for (int off = 16; off >= 1; off >>= 1)
        acc += __shfl_down(acc, off, 32);
    if (lane == 0) xp[ms * NXP + p] = acc;
}

// ====================================================================
// Sequential selective-scan, lane-per-state layout:
// each 16-lane half-group owns one (b,d) channel; lane n owns state n.
// Per step: 1 exp + 2 FMA per lane, then a 4-stage shfl_xor reduction
// inside the half-group for y = <state, C>. 98304 lanes = 3072 waves.
// ====================================================================
__global__ void scan_kernel(const float* __restrict__ xc,      // [M_TOT][DIN]
                            const float* __restrict__ xz,      // z in cols [DIN, 2*DIN)
                            const float* __restrict__ xp,      // [M_TOT][NXP]
                            const float* __restrict__ A_log,   // [DIN][NST]
                            const float* __restrict__ Dvec,    // [DIN]
                            float* __restrict__ y,             // [M_TOT][DIN]
                            float* __restrict__ state_out)     // [BSZ][DIN][NST]
{
    const int t = blockIdx.x * blockDim.x + threadIdx.x;       // over BSZ*DIN*NST
    const int n = t & (NST - 1);                               // state index
    const int g = t >> 4;                                      // group = b*DIN + d
    if (g >= BSZ * DIN) return;
    const int d = g % DIN;
    const int b = g / DIN;

    const float An = -expf(A_log[(size_t)d * NST + n]);
    const float Dd = Dvec[d];

    float st = 0.0f;
    for (int s = 0; s < SEQ; ++s) {
        const long long ms = (long long)b * SEQ + s;
        const float* __restrict__ xpr = xp + ms * NXP;
        const float dt = softplus_f(xpr[0]);
        const float xt = xc[ms * DIN + d];

        const float a = expf(dt * An);
        st = st * a + xt * xpr[1 + n];
        float yv = st * xpr[1 + NST + n];
        // reduce over the 16 lanes of this half-group (stays inside wave32)
        yv += __shfl_xor(yv, 1, 32);
        yv += __shfl_xor(yv, 2, 32);
        yv += __shfl_xor(yv, 4, 32);
        yv += __shfl_xor(yv, 8, 32);

        if (n == 0) {
            const float zt = xz[ms * NXZ + DIN + d];
            y[ms * DIN + d] = (yv + xt * Dd) * silu_f(zt);
        }
    }

    state_out[(size_t)g * NST + n] = st;   // fully coalesced
}

// ====================================================================
extern "C" void kernel_launch(void* const* d_in, const int* in_sizes, int n_in,
                              void* d_out, int out_size, void* d_ws, size_t ws_size,
                              hipStream_t stream)
{
    const float* x      = (const float*)d_in[0];   // [4,2048,768]
    const float* W_in   = (const float*)d_in[1];   // [3072,768]
    const float* conv_w = (const float*)d_in[2];   // [1536,1,4]
    const float* W_xp   = (const float*)d_in[3];   // [33,1536]
    const float* A_log  = (const float*)d_in[4];   // [1536,16]
    const float* Dv     = (const float*)d_in[5];   // [1536]
    const float* W_out  = (const float*)d_in[6];   // [768,1536]

    float* out = (float*)d_out;                          // [4,2048,768] flat
    float* state_out = out + (size_t)M_TOT * HID;        // [4,1536,16] flat

    // workspace layout (~202 MB)
    float* xz = (float*)d_ws;                            // M_TOT*NXZ
    float* xc = xz + (size_t)M_TOT * NXZ;                // M_TOT*DIN
    float* xp = xc + (size_t)M_TOT * DIN;                // M_TOT*NXP
    float* yb = xp + (size_t)M_TOT * NXP;                // M_TOT*DIN

    // 1) xz = x @ W_in^T   (M=8192, N=3072, K=768): 256*48 wave-tiles
    {
        const int wave_tiles = (M_TOT / 32) * (NXZ / 64);
        wmma_gemm_nt<HID, NXZ><<<wave_tiles / 8, 256, 0, stream>>>(x, W_in, xz);
    }

    // 2) depthwise causal conv + SiLU -> xc (float4 over channels)
    {
        const long long total = (long long)M_TOT * (DIN / 4);
        const int blocks = (int)((total + 255) / 256);
        conv_silu_kernel<<<blocks, 256, 0, stream>>>(xz, conv_w, xc);
    }

    // 3) xp = xc @ W_xp^T  (one wave per output scalar)
    {
        const long long waves = (long long)M_TOT * NXP;
        const int blocks = (int)((waves * 32 + 255) / 256);
        xp_kernel<<<blocks, 256, 0, stream>>>(xc, W_xp, xp);
    }

    // 4) sequential scan + gating -> yb, final state -> d_out tail
    {
        const int threads = BSZ * DIN * NST;             // 98304
        scan_kernel<<<threads / 256, 256, 0, stream>>>(xc, xz, xp, A_log, Dv,
                                                       yb, state_out);
    }

    // 5) out = yb @ W_out^T  (M=8192, N=768, K=1536): 256*12 wave-tiles
    {
        const int wave_tiles = (M_TOT / 32) * (HID / 64);
        wmma_gemm_nt<DIN, HID><<<wave_tiles / 8, 256, 0, stream>>>(yb, W_out, out);
    }
}